// Decoder_61735859913045
// MI455X (gfx1250) — compile-verified
//
#include <hip/hip_runtime.h>
#include <math.h>

typedef float v2f __attribute__((ext_vector_type(2)));
typedef float v8f __attribute__((ext_vector_type(8)));

#define B_      64
#define E_      128
#define H_      256
#define H2_     512
#define TK_     2048
#define V_      50000
#define NEXTRA_ 50
#define VEXT_   (V_ + NEXTRA_)

__device__ __forceinline__ float sigmf(float x) { return 1.0f / (1.0f + expf(-x)); }

// Async global->LDS b128 copy (CDNA5 GLOBAL_LOAD_ASYNC_TO_LDS_B128, ASYNCcnt-tracked).
// lds_addr: byte offset within workgroup LDS (addrspace(3) offset).
__device__ __forceinline__ void async_copy_b128(unsigned lds_addr, const void* gaddr) {
    asm volatile("global_load_async_to_lds_b128 %0, %1, off"
                 :: "v"(lds_addr), "v"(gaddr)
                 : "memory");
}

// ---------------------------------------------------------------------------
// K1: x = concat(c_t_1, emb[y]) @ W_xc.T + b_xc        -> ws_x [B,E]
// ---------------------------------------------------------------------------
__global__ void k_input_x(const int* __restrict__ y, const float* __restrict__ c_t_1,
                          const float* __restrict__ emb, const float* __restrict__ Wxc,
                          const float* __restrict__ bxc, float* __restrict__ x_out) {
    int i = blockIdx.x * blockDim.x + threadIdx.x;
    if (i >= B_ * E_) return;
    int b = i / E_, j = i % E_;
    const float* wr = Wxc + (long)j * (H2_ + E_);
    const float* c  = c_t_1 + (long)b * H2_;
    float acc = bxc[j];
    for (int k = 0; k < H2_; ++k) acc += c[k] * wr[k];
    const float* e = emb + (long)y[b] * E_;
    for (int k = 0; k < E_; ++k) acc += e[k] * wr[H2_ + k];
    x_out[i] = acc;
}

// ---------------------------------------------------------------------------
// K2: single-step LSTM (gate order i,f,g,o) -> h_new, c_new (into d_out)
// ---------------------------------------------------------------------------
__global__ void k_lstm(const float* __restrict__ x, const float* __restrict__ h0,
                       const float* __restrict__ c0, const float* __restrict__ Wih,
                       const float* __restrict__ bih, const float* __restrict__ Whh,
                       const float* __restrict__ bhh, float* __restrict__ h_new,
                       float* __restrict__ c_new) {
    int i = blockIdx.x * blockDim.x + threadIdx.x;
    if (i >= B_ * H_) return;
    int b = i / H_, j = i % H_;
    const float* xb = x + (long)b * E_;
    const float* hb = h0 + (long)b * H_;
    float g[4];
    for (int gi = 0; gi < 4; ++gi) {
        int row = gi * H_ + j;
        float acc = bih[row] + bhh[row];
        const float* wi = Wih + (long)row * E_;
        for (int k = 0; k < E_; ++k) acc += xb[k] * wi[k];
        const float* wh = Whh + (long)row * H_;
        for (int k = 0; k < H_; ++k) acc += hb[k] * wh[k];
        g[gi] = acc;
    }
    float cn = sigmf(g[1]) * c0[i] + sigmf(g[0]) * tanhf(g[2]);
    float hn = sigmf(g[3]) * tanhf(cn);
    h_new[i] = hn;
    c_new[i] = cn;
}

// ---------------------------------------------------------------------------
// K3: dec_fea = [h_new, c_new] @ W_dp.T + b_dp -> ws_dec [B,2H]
// ---------------------------------------------------------------------------
__global__ void k_decfea(const float* __restrict__ h_new, const float* __restrict__ c_new,
                         const float* __restrict__ Wdp, const float* __restrict__ bdp,
                         float* __restrict__ dec) {
    int i = blockIdx.x * blockDim.x + threadIdx.x;
    if (i >= B_ * H2_) return;
    int b = i / H2_, j = i % H2_;
    const float* w  = Wdp + (long)j * H2_;
    const float* hb = h_new + (long)b * H_;
    const float* cb = c_new + (long)b * H_;
    float acc = bdp[j];
    for (int k = 0; k < H_; ++k) acc += hb[k] * w[k];
    for (int k = 0; k < H_; ++k) acc += cb[k] * w[H_ + k];
    dec[i] = acc;
}

// ---------------------------------------------------------------------------
// K4: scores[b,t] = sum_n tanh(enc_feat + dec_fea + cov*Wc) * v   (256 MB stream)
// one wave32 per (b,t); float4 coalesced loads; shuffle reduction
// ---------------------------------------------------------------------------
__global__ void k_scores(const float* __restrict__ enc_feat, const float* __restrict__ dec,
                         const float* __restrict__ cov, const float* __restrict__ Wc,
                         const float* __restrict__ vw, float* __restrict__ scores) {
    int wid  = (blockIdx.x * blockDim.x + threadIdx.x) >> 5;
    int lane = threadIdx.x & 31;
    if (wid >= B_ * TK_) return;
    int b = wid / TK_;
    float cv = cov[wid];
    const float4* ef  = (const float4*)(enc_feat + (long)wid * H2_);
    const float4* df  = (const float4*)(dec + (long)b * H2_);
    const float4* wc4 = (const float4*)Wc;
    const float4* v4  = (const float4*)vw;
    float acc = 0.f;
    for (int i = lane; i < H2_ / 4; i += 32) {
        float4 e = ef[i], d = df[i], w = wc4[i], v = v4[i];
        acc += tanhf(e.x + d.x + cv * w.x) * v.x;
        acc += tanhf(e.y + d.y + cv * w.y) * v.y;
        acc += tanhf(e.z + d.z + cv * w.z) * v.z;
        acc += tanhf(e.w + d.w + cv * w.w) * v.w;
    }
    for (int off = 16; off > 0; off >>= 1) acc += __shfl_down(acc, off, 32);
    if (lane == 0) scores[wid] = acc;
}

// ---------------------------------------------------------------------------
// K5: attn = softmax(scores)*mask renormalized; coverage_next = cov + attn
// (folded: attn = exp(s-max)*mask / sum(exp(s-max)*mask))
// ---------------------------------------------------------------------------
__global__ void k_attn(const float* __restrict__ scores, const float* __restrict__ mask,
                       const float* __restrict__ cov, float* __restrict__ attn,
                       float* __restrict__ cov_next) {
    __shared__ float red[256];
    int b = blockIdx.x, tid = threadIdx.x;
    const float* s = scores + (long)b * TK_;
    const float* m = mask   + (long)b * TK_;
    float lmax = -1e30f;
    for (int t = tid; t < TK_; t += 256) lmax = fmaxf(lmax, s[t]);
    red[tid] = lmax; __syncthreads();
    for (int st = 128; st > 0; st >>= 1) { if (tid < st) red[tid] = fmaxf(red[tid], red[tid + st]); __syncthreads(); }
    float mx = red[0]; __syncthreads();
    float lsum = 0.f;
    for (int t = tid; t < TK_; t += 256) lsum += expf(s[t] - mx) * m[t];
    red[tid] = lsum; __syncthreads();
    for (int st = 128; st > 0; st >>= 1) { if (tid < st) red[tid] += red[tid + st]; __syncthreads(); }
    float inv = 1.0f / red[0];
    for (int t = tid; t < TK_; t += 256) {
        float a = expf(s[t] - mx) * m[t] * inv;
        attn[(long)b * TK_ + t]     = a;
        cov_next[(long)b * TK_ + t] = cov[(long)b * TK_ + t] + a;
    }
}

// ---------------------------------------------------------------------------
// K6: c_t[b,n] = sum_t attn[b,t] * enc_out[b,t,n]   (256 MB stream)
// Double-buffered GLOBAL_LOAD_ASYNC_TO_LDS_B128 streaming of 16x512-float tiles.
// Each wave issues 4 async b128 per tile; s_wait_asynccnt 4 after issuing the
// next tile guarantees the current tile landed (async loads complete in-order),
// then a workgroup barrier makes it visible to all 16 waves.
// ---------------------------------------------------------------------------
#define CT_TILE    16
#define CT_THREADS 512
#define CT_NT      (TK_ / CT_TILE)

__global__ void __launch_bounds__(CT_THREADS)
k_ctx(const float* __restrict__ attn, const float* __restrict__ enc_out,
      float* __restrict__ c_t) {
    __shared__ float sa[TK_];
    __shared__ float buf[2][CT_TILE * H2_];
    int b = blockIdx.x, tid = threadIdx.x;
    for (int t = tid; t < TK_; t += CT_THREADS) sa[t] = attn[(long)b * TK_ + t];

    const float* gbase = enc_out + (long)b * TK_ * H2_;

    auto issue_tile = [&](int tile, int bsel) {
        const float4* g = (const float4*)(gbase + (long)tile * CT_TILE * H2_);
        unsigned lbase = (unsigned)(uintptr_t)(&buf[bsel][0]);
#pragma unroll
        for (int c = 0; c < 4; ++c) {
            int idx = tid + c * CT_THREADS;            // 2048 float4 per tile
            async_copy_b128(lbase + (unsigned)idx * 16u, (const void*)(g + idx));
        }
    };

    issue_tile(0, 0);
    float acc = 0.f;
    for (int tt = 0; tt < CT_NT; ++tt) {
        int cur = tt & 1;
        if (tt + 1 < CT_NT) {
            issue_tile(tt + 1, cur ^ 1);
            asm volatile("s_wait_asynccnt 4" ::: "memory");
        } else {
            asm volatile("s_wait_asynccnt 0" ::: "memory");
        }
        __syncthreads();
        const float* bt  = &buf[cur][0];
        const float* sat = &sa[tt * CT_TILE];
#pragma unroll
        for (int r = 0; r < CT_TILE; ++r) acc += sat[r] * bt[r * H2_ + tid];
        __syncthreads();   // tile consumed; its buffer may be refilled next iter
    }
    c_t[(long)b * H2_ + tid] = acc;
}

// ---------------------------------------------------------------------------
// K7: p_gen = sigmoid([c_t, h_new, c_new, x] @ W_pg.T + b_pg); wave32 per b
// ---------------------------------------------------------------------------
__global__ void k_pgen(const float* __restrict__ c_t, const float* __restrict__ h_new,
                       const float* __restrict__ c_new, const float* __restrict__ x,
                       const float* __restrict__ Wpg, const float* __restrict__ bpg,
                       float* __restrict__ pgen) {
    int wid  = (blockIdx.x * blockDim.x + threadIdx.x) >> 5;
    int lane = threadIdx.x & 31;
    if (wid >= B_) return;
    int b = wid;
    float acc = 0.f;
    for (int j = lane; j < H2_; j += 32) acc += c_t[(long)b * H2_ + j] * Wpg[j];
    for (int j = lane; j < H_;  j += 32) acc += h_new[(long)b * H_ + j] * Wpg[H2_ + j];
    for (int j = lane; j < H_;  j += 32) acc += c_new[(long)b * H_ + j] * Wpg[H2_ + H_ + j];
    for (int j = lane; j < E_;  j += 32) acc += x[(long)b * E_ + j] * Wpg[H2_ + 2 * H_ + j];
    for (int off = 16; off > 0; off >>= 1) acc += __shfl_down(acc, off, 32);
    if (lane == 0) pgen[b] = sigmf(acc + bpg[0]);
}

// ---------------------------------------------------------------------------
// K8: out1 = [h_new, c_t] @ W_o1.T + b_o1 -> ws_out [B,H]
// ---------------------------------------------------------------------------
__global__ void k_out1(const float* __restrict__ h_new, const float* __restrict__ c_t,
                       const float* __restrict__ Wo1, const float* __restrict__ bo1,
                       float* __restrict__ out) {
    int i = blockIdx.x * blockDim.x + threadIdx.x;
    if (i >= B_ * H_) return;
    int b = i / H_, j = i % H_;
    const float* w  = Wo1 + (long)j * (3 * H_);
    const float* hb = h_new + (long)b * H_;
    const float* cb = c_t + (long)b * H2_;
    float acc = bo1[j];
    for (int k = 0; k < H_;  ++k) acc += hb[k] * w[k];
    for (int k = 0; k < H2_; ++k) acc += cb[k] * w[H_ + k];
    out[i] = acc;
}

// ---------------------------------------------------------------------------
// K9: logits = out1 @ W_o2.T + b_o2  via V_WMMA_F32_16X16X4_F32
// One wave owns a 64x16 output tile: 4 accumulators (M-tiles), K=256 -> 256 WMMAs.
// A 16x4 f32 frag: lanes 0-15 hold M=lane, K={k0,k0+1}; lanes 16-31 K={k0+2,k0+3}.
// B 4x16 f32 frag (mirrored): N=lane&15, same K split across lane halves.
// ---------------------------------------------------------------------------
__global__ void __launch_bounds__(256)
k_logits_wmma(const float* __restrict__ out, const float* __restrict__ Wo2,
              const float* __restrict__ bo2, float* __restrict__ logits) {
    int wid  = (blockIdx.x * blockDim.x + threadIdx.x) >> 5;
    int lane = threadIdx.x & 31;
    if (wid >= V_ / 16) return;
    int n0    = wid * 16;
    int nlane = lane & 15;
    int khalf = (lane >> 4) * 2;
    v8f acc0 = {}, acc1 = {}, acc2 = {}, acc3 = {};
    const float* brow = Wo2 + (long)(n0 + nlane) * H_;
    const float* ar0  = out + (long)(0 * 16 + nlane) * H_;
    const float* ar1  = out + (long)(1 * 16 + nlane) * H_;
    const float* ar2  = out + (long)(2 * 16 + nlane) * H_;
    const float* ar3  = out + (long)(3 * 16 + nlane) * H_;
    for (int k0 = 0; k0 < H_; k0 += 4) {
        int kb = k0 + khalf;
        v2f bf = *(const v2f*)(brow + kb);
        v2f a0 = *(const v2f*)(ar0 + kb);
        v2f a1 = *(const v2f*)(ar1 + kb);
        v2f a2 = *(const v2f*)(ar2 + kb);
        v2f a3 = *(const v2f*)(ar3 + kb);
        acc0 = __builtin_amdgcn_wmma_f32_16x16x4_f32(false, a0, false, bf, (short)0, acc0, false, false);
        acc1 = __builtin_amdgcn_wmma_f32_16x16x4_f32(false, a1, false, bf, (short)0, acc1, false, false);
        acc2 = __builtin_amdgcn_wmma_f32_16x16x4_f32(false, a2, false, bf, (short)0, acc2, false, false);
        acc3 = __builtin_amdgcn_wmma_f32_16x16x4_f32(false, a3, false, bf, (short)0, acc3, false, false);
    }
    float bias = bo2[n0 + nlane];
    int mhi = (lane >> 4) * 8;
    for (int r = 0; r < 8; ++r) {
        int m = mhi + r;
        logits[(long)(0 * 16 + m) * V_ + n0 + nlane] = acc0[r] + bias;
        logits[(long)(1 * 16 + m) * V_ + n0 + nlane] = acc1[r] + bias;
        logits[(long)(2 * 16 + m) * V_ + n0 + nlane] = acc2[r] + bias;
        logits[(long)(3 * 16 + m) * V_ + n0 + nlane] = acc3[r] + bias;
    }
}

// ---------------------------------------------------------------------------
// K10a: per-row max and sum(exp) for the vocab softmax
// ---------------------------------------------------------------------------
__global__ void k_rowstats(const float* __restrict__ logits, float* __restrict__ rowmax,
                           float* __restrict__ rowsum) {
    __shared__ float red[1024];
    int b = blockIdx.x, tid = threadIdx.x; // blockDim = 1024
    const float* l = logits + (long)b * V_;
    float lmax = -1e30f;
    for (int v = tid; v < V_; v += 1024) lmax = fmaxf(lmax, l[v]);
    red[tid] = lmax; __syncthreads();
    for (int st = 512; st > 0; st >>= 1) { if (tid < st) red[tid] = fmaxf(red[tid], red[tid + st]); __syncthreads(); }
    float m = red[0]; __syncthreads();
    float s = 0.f;
    for (int v = tid; v < V_; v += 1024) s += expf(l[v] - m);
    red[tid] = s; __syncthreads();
    for (int st = 512; st > 0; st >>= 1) { if (tid < st) red[tid] += red[tid + st]; __syncthreads(); }
    if (tid == 0) { rowmax[b] = m; rowsum[b] = red[0]; }
}

// ---------------------------------------------------------------------------
// K10b: final_dist[:, :V] = p_gen * softmax(logits); [V:] = extra_zeros
// ---------------------------------------------------------------------------
__global__ void k_finaldist(const float* __restrict__ logits, const float* __restrict__ rowmax,
                            const float* __restrict__ rowsum, const float* __restrict__ pgen,
                            const float* __restrict__ extra_zeros, float* __restrict__ fd) {
    long i = (long)blockIdx.x * blockDim.x + threadIdx.x;
    if (i >= (long)B_ * VEXT_) return;
    int b = (int)(i / VEXT_), v = (int)(i % VEXT_);
    float val;
    if (v < V_) val = pgen[b] * expf(logits[(long)b * V_ + v] - rowmax[b]) / rowsum[b];
    else        val = extra_zeros[(long)b * NEXTRA_ + (v - V_)];
    fd[i] = val;
}

// ---------------------------------------------------------------------------
// K11: scatter-add copy mass: fd[b, idx[b,t]] += (1-p_gen[b]) * attn[b,t]
// ---------------------------------------------------------------------------
__global__ void k_scatter(const float* __restrict__ attn, const float* __restrict__ pgen,
                          const int* __restrict__ ext_vocab, float* __restrict__ fd) {
    int i = blockIdx.x * blockDim.x + threadIdx.x;
    if (i >= B_ * TK_) return;
    int b = i / TK_;
    float add = (1.0f - pgen[b]) * attn[i];
    atomicAdd(&fd[(long)b * VEXT_ + ext_vocab[i]], add);
}

// ---------------------------------------------------------------------------
extern "C" void kernel_launch(void* const* d_in, const int* in_sizes, int n_in,
                              void* d_out, int out_size, void* d_ws, size_t ws_size,
                              hipStream_t stream) {
    (void)in_sizes; (void)n_in; (void)out_size; (void)ws_size;
    const int*   y_t_1     = (const int*)  d_in[0];
    const float* h_0       = (const float*)d_in[1];
    const float* c_0       = (const float*)d_in[2];
    const float* enc_out   = (const float*)d_in[3];
    const float* enc_feat  = (const float*)d_in[4];
    const float* pad_mask  = (const float*)d_in[5];
    const float* c_t_1     = (const float*)d_in[6];
    const float* extraz    = (const float*)d_in[7];
    const int*   ext_vocab = (const int*)  d_in[8];
    const float* coverage  = (const float*)d_in[9];
    // d_in[10] = step (unused)
    const float* emb   = (const float*)d_in[11];
    const float* Wxc   = (const float*)d_in[12];
    const float* bxc   = (const float*)d_in[13];
    const float* Wih   = (const float*)d_in[14];
    const float* bih   = (const float*)d_in[15];
    const float* Whh   = (const float*)d_in[16];
    const float* bhh   = (const float*)d_in[17];
    const float* Wdp   = (const float*)d_in[18];
    const float* bdp   = (const float*)d_in[19];
    const float* vw    = (const float*)d_in[20];
    const float* Wc    = (const float*)d_in[21];
    const float* Wpg   = (const float*)d_in[22];
    const float* bpg   = (const float*)d_in[23];
    const float* Wo1   = (const float*)d_in[24];
    const float* bo1   = (const float*)d_in[25];
    const float* Wo2   = (const float*)d_in[26];
    const float* bo2   = (const float*)d_in[27];

    // output layout (flat, return order)
    float* o = (float*)d_out;
    float* fd       = o;                                // [B, V+NEXTRA]
    float* h_new    = o + (long)B_ * VEXT_;             // [1,B,H]
    float* c_new    = h_new + (long)B_ * H_;            // [1,B,H]
    float* c_t      = c_new + (long)B_ * H_;            // [B,2H]
    float* attn     = c_t + (long)B_ * H2_;             // [B,TK]
    float* pgen     = attn + (long)B_ * TK_;            // [B,1]
    float* cov_next = pgen + B_;                        // [B,TK]

    // workspace layout (floats)
    float* w = (float*)d_ws;
    float* ws_x      = w;                               // [B,E]       8192
    float* ws_dec    = ws_x + (long)B_ * E_;            // [B,2H]      32768
    float* ws_scores = ws_dec + (long)B_ * H2_;         // [B,TK]      131072
    float* ws_out    = ws_scores + (long)B_ * TK_;      // [B,H]       16384
    float* ws_rowmax = ws_out + (long)B_ * H_;          // [B]
    float* ws_rowsum = ws_rowmax + B_;                  // [B]
    float* ws_logits = ws_rowsum + B_;                  // [B,V]       3.2M

    k_input_x<<<(B_ * E_ + 255) / 256, 256, 0, stream>>>(y_t_1, c_t_1, emb, Wxc, bxc, ws_x);
    k_lstm<<<(B_ * H_ + 255) / 256, 256, 0, stream>>>(ws_x, h_0, c_0, Wih, bih, Whh, bhh, h_new, c_new);
    k_decfea<<<(B_ * H2_ + 255) / 256, 256, 0, stream>>>(h_new, c_new, Wdp, bdp, ws_dec);
    k_scores<<<(B_ * TK_) / 8, 256, 0, stream>>>(enc_feat, ws_dec, coverage, Wc, vw, ws_scores);
    k_attn<<<B_, 256, 0, stream>>>(ws_scores, pad_mask, coverage, attn, cov_next);
    k_ctx<<<B_, CT_THREADS, 0, stream>>>(attn, enc_out, c_t);
    k_pgen<<<(B_ + 7) / 8, 256, 0, stream>>>(c_t, h_new, c_new, ws_x, Wpg, bpg, pgen);
    k_out1<<<(B_ * H_ + 255) / 256, 256, 0, stream>>>(h_new, c_t, Wo1, bo1, ws_out);
    k_logits_wmma<<<(V_ / 16 + 7) / 8, 256, 0, stream>>>(ws_out, Wo2, bo2, ws_logits);
    k_rowstats<<<B_, 1024, 0, stream>>>(ws_logits, ws_rowmax, ws_rowsum);
    k_finaldist<<<(int)(((long)B_ * VEXT_ + 255) / 256), 256, 0, stream>>>(ws_logits, ws_rowmax, ws_rowsum, pgen, extraz, fd);
    k_scatter<<<(B_ * TK_) / 256, 256, 0, stream>>>(attn, pgen, ext_vocab, fd);
}